// WaveletTransLayer_81587198755436
// MI455X (gfx1250) — compile-verified
//
#include <hip/hip_runtime.h>

// Fused symmetric-pad DWT analysis+synthesis (10-tap) for x:(64,128,4096) f32.
// ya[2m+r] = sum_{k=0}^{17} s[2m+k] * G_r[k],  G_r[k] = sum_u REC[k-2u]*REC[8+r-2u]
// computed per 16x16 output tile with chained V_WMMA_F32_16X16X4_F32 (Toeplitz B).

typedef float v2f __attribute__((ext_vector_type(2)));
typedef float v8f __attribute__((ext_vector_type(8)));

#define LSIG   524288          // 128*4096 flattened signal per batch
#define SEGS   256             // segments (blocks) per batch: 1024 m-pairs each
#define S_TILE 2064            // staged s-values per block (2048 + 16 halo)

struct GTab { float g[2][18]; };

__host__ __device__ constexpr GTab makeG() {
    constexpr float REC[10] = {
        0.160102397974125f, 0.6038292697974729f, 0.7243085284385744f,
        0.13842814590110342f, -0.24229488706619015f, -0.03224486958502952f,
        0.07757149384006515f, -0.006241490213011705f, -0.012580751999015526f,
        0.003335725285001549f };
    GTab t{};
    for (int r = 0; r < 2; ++r)
        for (int k = 0; k < 18; ++k) {
            float acc = 0.0f;
            for (int u = 0; u < 5; ++u) {
                int tt = k - 2 * u;
                if (tt >= 0 && tt <= 9) acc += REC[tt] * REC[8 + r - 2 * u];
            }
            t.g[r][k] = acc;
        }
    return t;
}

__device__ __constant__ GTab cG = makeG();

extern "C" __global__ __launch_bounds__(256)
void wavelet_fused_wmma(const float* __restrict__ x, float* __restrict__ y)
{
    __shared__ __align__(16) float s_tile[S_TILE];

    const int tid = threadIdx.x;
    const int blk = blockIdx.x;
    const int b   = blk >> 8;        // batch
    const int seg = blk & (SEGS - 1);
    const int S0  = seg << 11;       // s-index base = 2*m0_block = 2048*seg

    const float* __restrict__ xb = x + (size_t)b * LSIG;

    // ---- Stage signal tile (with 18-tap halo) into LDS: one HBM read per byte
    if (seg > 0 && seg < SEGS - 1) {
        // interior: S0-8 is 8-float (32B) aligned -> coalesced float4 copy
        const float4* __restrict__ src = (const float4*)(xb + S0 - 8);
        float4* dst = (float4*)s_tile;
        #pragma unroll 2
        for (int t = tid; t < S_TILE / 4; t += 256) dst[t] = src[t];
    } else {
        // edge: symmetric reflection s[i] = sig[i-8] with mirror at 0 and LSIG
        for (int t = tid; t < S_TILE; t += 256) {
            int idx = S0 + t - 8;
            if (idx < 0) idx = -idx - 1;
            else if (idx >= LSIG) idx = 2 * LSIG - idx - 1;
            s_tile[t] = xb[idx];
        }
    }
    __syncthreads();

    const int lane = tid & 31;
    const int wv   = tid >> 5;       // 8 waves per block, one 16x16 tile each
    const int half = lane >> 4;      // lane-half selects K pair (A/B) / M+8 (C)
    const int im   = lane & 15;      // A row (M) == C column (N)

    // ---- B fragments: Toeplitz of the two fused 18-tap filters (per-lane consts)
    // B[k][j] = G_{j&1}[k - (j&~1)], lane holds column j=im, K = 2*half + {0,1}
    const int r  = im & 1;
    const int jc = im & ~1;
    v2f bfrag[8];
    #pragma unroll
    for (int q = 0; q < 8; ++q) {
        #pragma unroll
        for (int kk = 0; kk < 2; ++kk) {
            int gi = 4 * q + 2 * half + kk - jc;
            bfrag[q][kk] = (gi >= 0 && gi < 18) ? cG.g[r][gi] : 0.0f;
        }
    }

    // ---- A fragments from LDS + 8 chained fp32 WMMAs (K = 0..31)
    // A[i][k] = s_tile[256*wv + 16*i + k]; lane: i=im, k = 4q + 2*half + {0,1}
    const int abase = (wv << 8) + (im << 4) + (half << 1);
    v8f c = {};
    #pragma unroll
    for (int q = 0; q < 8; ++q) {
        const v2f a = *(const v2f*)(s_tile + abase + 4 * q);
        c = __builtin_amdgcn_wmma_f32_16x16x4_f32(
                /*neg_a=*/false, a, /*neg_b=*/false, bfrag[q],
                /*c_mod=*/(short)0, c, /*reuse_a=*/false, /*reuse_b=*/false);
    }

    // ---- Store: C[i][j] -> ya[2*m0w + 16*i + j], 256 contiguous outputs/wave
    // lane l, VGPR g holds (M = g + 8*half, N = im)
    float* __restrict__ yb = y + (size_t)b * LSIG
                           + ((size_t)seg << 11) + (wv << 8) + (half << 7) + im;
    #pragma unroll
    for (int g = 0; g < 8; ++g)
        yb[g << 4] = c[g];
}

extern "C" void kernel_launch(void* const* d_in, const int* in_sizes, int n_in,
                              void* d_out, int out_size, void* d_ws, size_t ws_size,
                              hipStream_t stream) {
    const float* x = (const float*)d_in[0];
    float*       y = (float*)d_out;
    // 64 batches * 256 segments; each block emits 2048 contiguous fp32 outputs
    wavelet_fused_wmma<<<dim3(64 * SEGS), dim3(256), 0, stream>>>(x, y);
}